// GCN_52656299049581
// MI455X (gfx1250) — compile-verified
//
#include <hip/hip_runtime.h>
#include <hip/hip_bf16.h>
#include <math.h>

typedef float v2f __attribute__((ext_vector_type(2)));
typedef float v8f __attribute__((ext_vector_type(8)));

#define F_IN 256
#define H_DIM 128
#define BM 64
#define BN 128
#define KC 32

// ---------------- degree / norm ----------------
__global__ void deg_init(float* deg, int N) {
    int i = blockIdx.x * blockDim.x + threadIdx.x;
    if (i < N) deg[i] = 1.0f;  // self-loop weight
}

__global__ void deg_acc(const int* __restrict__ dst, const float* __restrict__ w,
                        float* deg, int E) {
    int e = blockIdx.x * blockDim.x + threadIdx.x;
    if (e < E) atomicAdd(&deg[dst[e]], w[e]);
}

__global__ void deg_rsqrt(float* deg, int N) {
    int i = blockIdx.x * blockDim.x + threadIdx.x;
    if (i < N) {
        float d = deg[i];
        deg[i] = (d > 0.0f) ? rsqrtf(d) : 0.0f;  // becomes dinv[]
    }
}

// ---------------- xw = x @ W^T via V_WMMA_F32_16X16X4_F32 ----------------
__global__ __launch_bounds__(256) void gemm_xw(const float* __restrict__ x,
                                               const float* __restrict__ W,
                                               float* __restrict__ xw, int N) {
    __shared__ float sX[BM][KC + 1];
    __shared__ float sW[BN][KC + 1];

    const int tid  = threadIdx.x;
    const int wave = tid >> 5;
    const int lane = tid & 31;
    const int m0   = blockIdx.x * BM;

    const int rt   = wave & 3;        // row tile 0..3  (16 rows each)
    const int cg   = wave >> 2;       // col group 0..1 (4 col-tiles each)
    const int lrow = lane & 15;
    const int koff = (lane >> 4) << 1;   // ISA A/B frag: lanes 16-31 hold K+2,K+3
    const int mhalf = (lane >> 4) << 3;  // C/D frag: lanes 16-31 hold rows +8

    v8f acc[4] = {};

    for (int k0 = 0; k0 < F_IN; k0 += KC) {
        // stage x tile (coalesced, zero-padded past N)
        for (int i = tid; i < BM * KC; i += 256) {
            int r = i >> 5, c = i & 31;
            int gr = m0 + r;
            sX[r][c] = (gr < N) ? x[(size_t)gr * F_IN + k0 + c] : 0.0f;
        }
        // stage W tile (coalesced; W is [128 x 256] row-major)
        for (int i = tid; i < BN * KC; i += 256) {
            int r = i >> 5, c = i & 31;
            sW[r][c] = W[(size_t)r * F_IN + k0 + c];
        }
        __syncthreads();

        #pragma unroll
        for (int kk = 0; kk < KC; kk += 4) {
            v2f a;
            a.x = sX[rt * 16 + lrow][kk + koff];
            a.y = sX[rt * 16 + lrow][kk + koff + 1];
            #pragma unroll
            for (int t = 0; t < 4; ++t) {
                int cn = (cg * 4 + t) * 16 + lrow;
                v2f bf;
                bf.x = sW[cn][kk + koff];
                bf.y = sW[cn][kk + koff + 1];
                acc[t] = __builtin_amdgcn_wmma_f32_16x16x4_f32(
                    false, a, false, bf, (short)0, acc[t], false, false);
            }
        }
        __syncthreads();
    }

    // store D: lane 0-15 -> rows v, cols lane; lane 16-31 -> rows v+8
    #pragma unroll
    for (int t = 0; t < 4; ++t) {
        int cn0 = (cg * 4 + t) * 16;
        #pragma unroll
        for (int v = 0; v < 8; ++v) {
            int gm = m0 + rt * 16 + v + mhalf;
            if (gm < N) xw[(size_t)gm * H_DIM + cn0 + lrow] = acc[t][v];
        }
    }
}

// ---------------- hacc = b + dinv^2 * xw  (bias + self-loop message) ----------------
__global__ void hinit(const float* __restrict__ xw, const float* __restrict__ dinv,
                      const float* __restrict__ b, float* __restrict__ hacc, int N) {
    size_t i = (size_t)blockIdx.x * blockDim.x + threadIdx.x;
    if (i >= (size_t)N * H_DIM) return;
    int n = (int)(i >> 7);
    int h = (int)(i & (H_DIM - 1));
    float di = dinv[n];
    hacc[i] = b[h] + di * di * xw[i];
}

// ---------------- edge scatter: one wave per edge ----------------
__global__ __launch_bounds__(256) void edge_scatter(const int* __restrict__ src,
                                                    const int* __restrict__ dst,
                                                    const float* __restrict__ w,
                                                    const float* __restrict__ dinv,
                                                    const float* __restrict__ xw,
                                                    float* __restrict__ hacc, int E) {
    int lane = threadIdx.x & 31;
    int e = (int)(((size_t)blockIdx.x * blockDim.x + threadIdx.x) >> 5);
    if (e >= E) return;
    int s = src[e], d = dst[e];
    float coef = dinv[s] * w[e] * dinv[d];
    const float4* xr = (const float4*)(xw + (size_t)s * H_DIM);
    float4 v = xr[lane];                        // 512B coalesced gather per wave
    float* hr = hacc + (size_t)d * H_DIM + lane * 4;
    atomicAdd(hr + 0, coef * v.x);
    atomicAdd(hr + 1, coef * v.y);
    atomicAdd(hr + 2, coef * v.z);
    atomicAdd(hr + 3, coef * v.w);
}

// ---------------- head: sigmoid(relu(h) @ W_lin^T + b_lin), one wave per node ----------------
__global__ __launch_bounds__(256) void head(const float* __restrict__ hacc,
                                            const float* __restrict__ W_lin,
                                            const float* __restrict__ b_lin,
                                            float* __restrict__ out, int N) {
    int lane = threadIdx.x & 31;
    int node = (int)(((size_t)blockIdx.x * blockDim.x + threadIdx.x) >> 5);
    if (node >= N) return;
    const float4* hr = (const float4*)(hacc + (size_t)node * H_DIM);
    float4 v = hr[lane];
    v.x = fmaxf(v.x, 0.0f); v.y = fmaxf(v.y, 0.0f);
    v.z = fmaxf(v.z, 0.0f); v.w = fmaxf(v.w, 0.0f);
    const float4* w0 = (const float4*)(W_lin);
    const float4* w1 = (const float4*)(W_lin + H_DIM);
    float4 a = w0[lane], c = w1[lane];
    float s0 = v.x * a.x + v.y * a.y + v.z * a.z + v.w * a.w;
    float s1 = v.x * c.x + v.y * c.y + v.z * c.z + v.w * c.w;
    #pragma unroll
    for (int off = 16; off > 0; off >>= 1) {
        s0 += __shfl_down(s0, off);
        s1 += __shfl_down(s1, off);
    }
    if (lane == 0) {
        float o0 = 1.0f / (1.0f + expf(-(s0 + b_lin[0])));
        float o1 = 1.0f / (1.0f + expf(-(s1 + b_lin[1])));
        out[(size_t)node * 2 + 0] = o0;
        out[(size_t)node * 2 + 1] = o1;
    }
}

extern "C" void kernel_launch(void* const* d_in, const int* in_sizes, int n_in,
                              void* d_out, int out_size, void* d_ws, size_t ws_size,
                              hipStream_t stream) {
    const float* x     = (const float*)d_in[0];
    const int*   ei    = (const int*)d_in[1];   // int32 (jax default, x64 disabled)
    const float* ew    = (const float*)d_in[2];
    const float* W     = (const float*)d_in[3];
    const float* b     = (const float*)d_in[4];
    const float* W_lin = (const float*)d_in[5];
    const float* b_lin = (const float*)d_in[6];

    const int N = in_sizes[0] / F_IN;
    const int E = in_sizes[1] / 2;
    const int* src = ei;
    const int* dst = ei + E;

    float* deg  = (float*)d_ws;                          // N floats -> becomes dinv
    float* xw   = deg + (((size_t)N + 255) & ~(size_t)255);
    float* hacc = xw + (size_t)N * H_DIM;
    float* out  = (float*)d_out;

    deg_init  <<<(N + 255) / 256, 256, 0, stream>>>(deg, N);
    deg_acc   <<<(E + 255) / 256, 256, 0, stream>>>(dst, ew, deg, E);
    deg_rsqrt <<<(N + 255) / 256, 256, 0, stream>>>(deg, N);

    gemm_xw   <<<(N + BM - 1) / BM, 256, 0, stream>>>(x, W, xw, N);

    hinit     <<<(int)(((size_t)N * H_DIM + 255) / 256), 256, 0, stream>>>(xw, deg, b, hacc, N);
    edge_scatter<<<(E + 7) / 8, 256, 0, stream>>>(src, dst, ew, deg, xw, hacc, E);
    head      <<<(N + 7) / 8, 256, 0, stream>>>(hacc, W_lin, b_lin, out, N);
}